// VectorQuantizer_80607946211815
// MI455X (gfx1250) — compile-verified
//
#include <hip/hip_runtime.h>
#include <math.h>
#include <stdint.h>

typedef float v2f __attribute__((ext_vector_type(2)));
typedef float v4f __attribute__((ext_vector_type(4)));
typedef float v8f __attribute__((ext_vector_type(8)));

#define NUM_EMB 512
#define EMB_DIM 64
#define HW      64                  // H == W == 64
#define NBATCH  32
#define N_ROWS  (NBATCH*HW*HW)      // 131072
#define LDST    68                  // padded LDS row stride (conflict-free b64 frag reads)
#define CHUNK   128                 // codes staged in LDS at a time
#define NCHUNK  (NUM_EMB/CHUNK)     // 4

// out layout (floats): [0]=loss, [1 .. 8388608]=quantized NCHW, [8388609]=perplexity,
// [8388610 ..]=encodings [131072,512]
#define Q_OFF    ((size_t)1)
#define PERP_OFF ((size_t)8388609)
#define ENC_OFF  ((size_t)8388610)

// ---- CDNA5 async global->LDS copy (ASYNCcnt-tracked), inline asm path ----
__device__ __forceinline__ void async_ld_b128(uint32_t lds_off, const void* gaddr) {
  // per-lane: copies 16 bytes from global address gaddr to LDS offset lds_off
  asm volatile("global_load_async_to_lds_b128 %0, %1, off"
               :: "v"(lds_off), "v"(gaddr) : "memory");
}
#define WAIT_ASYNCCNT(N) asm volatile("s_wait_asynccnt " #N ::: "memory")

__global__ void vq_init(unsigned int* hist) {
  hist[blockIdx.x * 256 + threadIdx.x] = 0u;
}

__global__ __launch_bounds__(256) void vq_main(const float* __restrict__ x_g,
                                               const float* __restrict__ w_g,
                                               float* __restrict__ out,
                                               float* __restrict__ loss_part,
                                               unsigned int* __restrict__ hist) {
  __shared__ __align__(16) float xs  [HW * LDST];        // x tile: [w][c]
  __shared__ __align__(16) float wbuf[2][CHUNK * LDST];  // double-buffered weight chunks
  __shared__ float nrm2[NUM_EMB];
  __shared__ float pmin[2][HW];
  __shared__ int   pidx[2][HW];
  __shared__ int   idxf[HW];
  __shared__ float lred[8];

  const int g    = blockIdx.x;        // 0..2047  -> (b,h)
  const int b    = g >> 6;
  const int h    = g & 63;
  const int t    = threadIdx.x;
  const int lane = t & 31;
  const int wave = t >> 5;

  // ---- kick off async staging of weight chunk 0 into wbuf[0] ----
  {
    #pragma unroll
    for (int i = 0; i < 8; ++i) {
      int id = wave * 256 + i * 32 + lane;   // float4 slot: 2048 per chunk
      int k = id >> 4, f4 = id & 15;
      uint32_t loff = (uint32_t)(uintptr_t)&wbuf[0][k * LDST + f4 * 4];
      async_ld_b128(loff, (const void*)(w_g + (size_t)k * EMB_DIM + f4 * 4));
    }
  }

  // ---- stage x tile while chunk 0 is in flight: xs[w][c] = inputs[b, c, h, w] ----
  const float* xbase = x_g + (size_t)b * (EMB_DIM * HW * HW) + (size_t)h * HW;
  #pragma unroll
  for (int i = 0; i < 16; ++i) {
    int l = t + 256 * i;              // 4096 elements
    int c = l >> 6, w = l & 63;
    xs[w * LDST + c] = __builtin_nontemporal_load(&xbase[(size_t)c * (HW * HW) + w]);
  }
  // ---- ||e||^2 for all 512 codes, straight from global (L2-resident) ----
  #pragma unroll
  for (int r = t; r < NUM_EMB; r += 256) {
    const v4f* wr = (const v4f*)(w_g + (size_t)r * EMB_DIM);
    float s = 0.f;
    #pragma unroll
    for (int d4 = 0; d4 < 16; ++d4) {
      v4f e = wr[d4];
      s = fmaf(e.x, e.x, s); s = fmaf(e.y, e.y, s);
      s = fmaf(e.z, e.z, s); s = fmaf(e.w, e.w, s);
    }
    nrm2[r] = s;
  }

  // ---- per-lane running argmin state ----
  float m[8]; int mi[8];
  #pragma unroll
  for (int v = 0; v < 8; ++v) { m[v] = 3.4e38f; mi[v] = NUM_EMB; }

  const int rt   = wave & 3;          // row tile (16 rows each)
  const int par  = wave >> 2;         // code-tile parity
  const int half = lane >> 4;
  const int ln   = lane & 15;
  const int arow = rt * 16 + ln;      // A-fragment row for this lane

  for (int chunk = 0; chunk < NCHUNK; ++chunk) {
    const int bsel = chunk & 1;
    if (chunk < NCHUNK - 1) {
      // prefetch next chunk into the other buffer (its prior contents were
      // released by the end-of-iteration barrier of chunk-1)
      const float* wn = w_g + (size_t)(chunk + 1) * CHUNK * EMB_DIM;
      #pragma unroll
      for (int i = 0; i < 8; ++i) {
        int id = wave * 256 + i * 32 + lane;
        int k = id >> 4, f4 = id & 15;
        uint32_t loff = (uint32_t)(uintptr_t)&wbuf[bsel ^ 1][k * LDST + f4 * 4];
        async_ld_b128(loff, (const void*)(wn + (size_t)k * EMB_DIM + f4 * 4));
      }
      WAIT_ASYNCCNT(8);   // in-order completion: current chunk's 8 loads done
    } else {
      WAIT_ASYNCCNT(0);   // last chunk fully arrived
    }
    __syncthreads();      // whole chunk visible to all waves (also covers xs/nrm2)

    const float* wsm = wbuf[bsel];
    // 8 local 16-code tiles; this wave handles 4 (its parity)
    #pragma unroll
    for (int j = 0; j < 4; ++j) {
      int ctl  = par + 2 * j;         // local code tile 0..7
      int crow = ctl * 16 + ln;       // this lane's code row within chunk
      v8f acc = {0.f, 0.f, 0.f, 0.f, 0.f, 0.f, 0.f, 0.f};
      #pragma unroll
      for (int kk = 0; kk < 16; ++kk) {  // K = 64 in steps of 4
        v2f a  = *(const v2f*)&xs [arow * LDST + 4 * kk + 2 * half];
        v2f bf = *(const v2f*)&wsm[crow * LDST + 4 * kk + 2 * half];
        acc = __builtin_amdgcn_wmma_f32_16x16x4_f32(false, a, false, bf,
                                                    (short)0, acc, false, false);
      }
      int   kidx = chunk * CHUNK + crow;   // global code index of this lane's column
      float nv   = nrm2[kidx];
      #pragma unroll
      for (int v = 0; v < 8; ++v) {
        float s = fmaf(-2.f, acc[v], nv);  // ||e||^2 - 2 x.e  (||x||^2 constant per row)
        if (s < m[v] || (s == m[v] && kidx < mi[v])) { m[v] = s; mi[v] = kidx; }
      }
    }
    __syncthreads();      // all waves done reading wbuf[bsel] before it is re-filled
  }

  // ---- argmin reduce across the 16 lanes sharing each row ----
  #pragma unroll
  for (int v = 0; v < 8; ++v) {
    #pragma unroll
    for (int off = 1; off < 16; off <<= 1) {
      float om = __shfl_xor(m[v],  off, 32);
      int   oi = __shfl_xor(mi[v], off, 32);
      if (om < m[v] || (om == m[v] && oi < mi[v])) { m[v] = om; mi[v] = oi; }
    }
    if (ln == 0) {
      int row = rt * 16 + v + 8 * half;
      pmin[par][row] = m[v];
      pidx[par][row] = mi[v];
    }
  }
  __syncthreads();

  if (t < HW) {                       // combine the two parities; histogram
    float m0 = pmin[0][t], m1 = pmin[1][t];
    int   i0 = pidx[0][t], i1 = pidx[1][t];
    int win = (m1 < m0 || (m1 == m0 && i1 < i0)) ? i1 : i0;
    idxf[t] = win;
    atomicAdd(&hist[win], 1u);
  }
  __syncthreads();

  // ---- gather quantized rows into wbuf[0] (coalesced along code row) ----
  float* qs = wbuf[0];
  #pragma unroll
  for (int i = 0; i < 16; ++i) {
    int l = t + 256 * i;
    int w = l >> 6, c = l & 63;
    qs[w * LDST + c] = w_g[(size_t)idxf[w] * EMB_DIM + c];
  }
  __syncthreads();

  // ---- quantized NCHW (NT stores) + loss partial ----
  float facc = 0.f;
  float* qout = out + Q_OFF + (size_t)b * (EMB_DIM * HW * HW) + (size_t)h * HW;
  #pragma unroll
  for (int i = 0; i < 16; ++i) {
    int l = t + 256 * i;
    int c = l >> 6, w = l & 63;      // consecutive t -> consecutive w: coalesced
    float q = qs[w * LDST + c];
    float d = q - xs[w * LDST + c];
    facc = fmaf(d, d, facc);
    __builtin_nontemporal_store(q, &qout[(size_t)c * (HW * HW) + w]);
  }

  // ---- one-hot encodings, 64-bit NT stores (segment base only 8B aligned) ----
  float* enc = out + ENC_OFF + (size_t)g * HW * NUM_EMB;
  for (int i = 0; i < 64; ++i) {
    int l  = t + 256 * i;            // 16384 v2f slots (64 rows x 256)
    int w  = l >> 8, c2 = l & 255;
    int wi = idxf[w];
    v2f v = {0.f, 0.f};
    if ((wi >> 1) == c2) v[wi & 1] = 1.0f;
    __builtin_nontemporal_store(v, (v2f*)&enc[(size_t)w * NUM_EMB + c2 * 2]);
  }

  // ---- deterministic per-WG loss partial ----
  #pragma unroll
  for (int off = 16; off > 0; off >>= 1) facc += __shfl_xor(facc, off, 32);
  if (lane == 0) lred[wave] = facc;
  __syncthreads();
  if (t == 0) {
    float s = 0.f;
    #pragma unroll
    for (int i = 0; i < 8; ++i) s += lred[i];
    loss_part[g] = s;
  }
}

__global__ __launch_bounds__(256) void vq_final(const float* __restrict__ loss_part,
                                                const unsigned int* __restrict__ hist,
                                                float* __restrict__ out) {
  __shared__ float red[256];
  int t = threadIdx.x;
  float ls = 0.f;
  #pragma unroll
  for (int i = 0; i < 8; ++i) ls += loss_part[t + 256 * i];   // 2048 partials, fixed order
  float ps = 0.f;
  #pragma unroll
  for (int i = 0; i < 2; ++i) {
    float p = (float)hist[t + 256 * i] * (1.0f / (float)N_ROWS);
    ps += p * logf(p + 1e-10f);
  }
  red[t] = ls; __syncthreads();
  for (int s = 128; s > 0; s >>= 1) { if (t < s) red[t] += red[t + s]; __syncthreads(); }
  float loss = 1.25f * red[0] / 8388608.0f;   // (1 + 0.25) * mean((q-x)^2)
  __syncthreads();
  red[t] = ps; __syncthreads();
  for (int s = 128; s > 0; s >>= 1) { if (t < s) red[t] += red[t + s]; __syncthreads(); }
  if (t == 0) {
    out[0]        = loss;
    out[PERP_OFF] = expf(-red[0]);
  }
}

extern "C" void kernel_launch(void* const* d_in, const int* in_sizes, int n_in,
                              void* d_out, int out_size, void* d_ws, size_t ws_size,
                              hipStream_t stream) {
  const float* x = (const float*)d_in[0];   // inputs [32,64,64,64] f32
  const float* w = (const float*)d_in[1];   // weight [512,64] f32
  float* out = (float*)d_out;

  unsigned int* hist = (unsigned int*)d_ws;                          // 512 u32
  float* loss_part   = (float*)((char*)d_ws + NUM_EMB * sizeof(unsigned int)); // 2048 f32

  vq_init <<<2,    256, 0, stream>>>(hist);
  vq_main <<<2048, 256, 0, stream>>>(x, w, out, loss_part, hist);
  vq_final<<<1,    256, 0, stream>>>(loss_part, hist, out);
}